// NaiveGNN_17351667876072
// MI455X (gfx1250) — compile-verified
//
#include <hip/hip_runtime.h>
#include <hip/hip_bf16.h>

typedef __attribute__((ext_vector_type(2))) float v2f;
typedef __attribute__((ext_vector_type(8))) float v8f;

#define N_CELL 100000
#define N_NET  100000
#define N_PIN  500000
#define N_NN   200000
#define N_PAIR 500000

// ---------------------------------------------------------------------------
// V_WMMA_F32_16X16X4_F32 : D(16x16 f32) = A(16x4 f32) * B(4x16 f32) + C
// A frag (2 VGPR): lane L -> M=L%16, K = (L/16)*2 + v
// B frag (2 VGPR): lane L -> N=L%16, K = (L/16)*2 + v
// C/D (8 VGPR):    lane L -> N=L%16, M = r + (L/16)*8
// ---------------------------------------------------------------------------
__device__ __forceinline__ v8f wmma_f32_k4(v2f a, v2f b, v8f c) {
  return __builtin_amdgcn_wmma_f32_16x16x4_f32(false, a, false, b, (short)0, c,
                                               false, false);
}

// ---------------------------------------------------------------------------
// CDNA5 async global->LDS copy (ASYNCcnt-tracked DMA, no VGPR round trip).
// Generic pointers to __shared__ carry the wave-relative LDS byte offset in
// their low 32 bits (ISA 10.2); the instruction adds
// LDS_BASE itself, so truncation gives the correct VDST operand.
// ---------------------------------------------------------------------------
__device__ __forceinline__ void async_g2l_b128(void* lds_dst, const void* gsrc) {
  unsigned l = (unsigned)(unsigned long long)lds_dst;
  unsigned long long g = (unsigned long long)gsrc;
  asm volatile("global_load_async_to_lds_b128 %0, %1, off"
               :: "v"(l), "v"(g)
               : "memory");
}
__device__ __forceinline__ void wait_async0() {
  asm volatile("s_wait_asynccnt 0x0" ::: "memory");
}

__device__ __forceinline__ float softplusf(float x) {
  return fmaxf(x, 0.0f) + log1pf(expf(-fabsf(x)));
}
__device__ __forceinline__ float sspf(float x) {  // DGL ShiftedSoftplus
  return softplusf(x) - 0.69314718055994530942f;
}

enum { ACT_NONE = 0, ACT_TANH = 1, ACT_SSP = 2 };
template <int ACT> __device__ __forceinline__ float actf(float x) {
  if (ACT == ACT_TANH) return tanhf(x);
  if (ACT == ACT_SSP) return sspf(x);
  return x;
}

// ---------------------------------------------------------------------------
__global__ void k_zero(float* __restrict__ p, long long n) {
  long long i = (long long)blockIdx.x * blockDim.x + threadIdx.x;
  long long stride = (long long)gridDim.x * blockDim.x;
  for (; i < n; i += stride) p[i] = 0.0f;
}

__global__ void k_count(const int* __restrict__ idx, float* __restrict__ deg,
                        int n) {
  int i = blockIdx.x * blockDim.x + threadIdx.x;
  if (i < n) atomicAdd(&deg[idx[i]], 1.0f);
}

__global__ void k_invsqrt_clip(float* __restrict__ d, int n) {
  int i = blockIdx.x * blockDim.x + threadIdx.x;
  if (i < n) d[i] = rsqrtf(fmaxf(d[i], 1.0f));
}

// ---------------------------------------------------------------------------
// Y[r0..r0+15, 0..63] = tanh( X[16x16] @ W[16x64] + b ) ; block=128 (4 waves)
__global__ void k_embed16(const float* __restrict__ X, const float* __restrict__ W,
                          const float* __restrict__ bias, float* __restrict__ Y) {
  __shared__ __align__(16) float sx[16 * 16];
  const int t = threadIdx.x;
  const int lane = t & 31, wv = t >> 5, hf = lane >> 4, lm = lane & 15;
  const int r0 = blockIdx.x * 16;
  const int col = wv * 16 + lm;
  if (t < 64) async_g2l_b128(sx + t * 4, X + (long long)r0 * 16 + t * 4);
  wait_async0();
  __syncthreads();
  v8f c = {};
#pragma unroll
  for (int kb = 0; kb < 16; kb += 4) {
    const int kk = kb + hf * 2;
    v2f a = *(const v2f*)(sx + lm * 16 + kk);
    v2f b;
    b.x = W[(kk + 0) * 64 + col];
    b.y = W[(kk + 1) * 64 + col];
    c = wmma_f32_k4(a, b, c);
  }
  const float bb = bias[col];
#pragma unroll
  for (int r = 0; r < 8; ++r) {
    const int row = r + hf * 8;
    Y[(long long)(r0 + row) * 64 + col] = tanhf(c[r] + bb);
  }
}

// h_pin = tanh(pin_feat[Px8] @ W[8x8] + b) : one thread per pin
__global__ void k_pin_embed(const float* __restrict__ X, const float* __restrict__ W,
                            const float* __restrict__ bias, float* __restrict__ Y,
                            int n) {
  int i = blockIdx.x * blockDim.x + threadIdx.x;
  if (i >= n) return;
  float x[8];
#pragma unroll
  for (int k = 0; k < 8; ++k) x[k] = X[(long long)i * 8 + k];
#pragma unroll
  for (int j = 0; j < 8; ++j) {
    float s = bias[j];
#pragma unroll
    for (int k = 0; k < 8; ++k) s += x[k] * W[k * 8 + j];
    Y[(long long)i * 8 + j] = tanhf(s);
  }
}

// agg[dst] += h[src] * cs[src] : one thread per (edge, feature)
__global__ void k_scatter_norm(const float* __restrict__ h, const int* __restrict__ src,
                               const int* __restrict__ dst, const float* __restrict__ cs,
                               float* __restrict__ agg, int nedge) {
  long long g = (long long)blockIdx.x * blockDim.x + threadIdx.x;
  if (g >= (long long)nedge * 64) return;
  int e = (int)(g >> 6);
  int f = (int)(g & 63);
  int s = src[e], d = dst[e];
  atomicAdd(&agg[(long long)d * 64 + f], h[(long long)s * 64 + f] * cs[s]);
}

// Y = act( X[Nx64] @ W[64x64] + b ) with async-LDS-staged A tiles; block=128
template <int ACT>
__global__ void k_gemm64(const float* __restrict__ X, const float* __restrict__ W,
                         const float* __restrict__ bias, float* __restrict__ Y) {
  __shared__ __align__(16) float sx[16 * 64];
  const int t = threadIdx.x;
  const int r0 = blockIdx.x * 16;
  const float* srcp = X + (long long)r0 * 64;
  async_g2l_b128(sx + t * 4, srcp + t * 4);
  async_g2l_b128(sx + 512 + t * 4, srcp + 512 + t * 4);
  wait_async0();
  __syncthreads();
  const int lane = t & 31, wv = t >> 5, hf = lane >> 4, lm = lane & 15;
  const int col = wv * 16 + lm;
  v8f c = {};
#pragma unroll 4
  for (int kb = 0; kb < 64; kb += 4) {
    const int kk = kb + hf * 2;
    v2f a = *(const v2f*)(sx + lm * 64 + kk);
    v2f b;
    b.x = W[(kk + 0) * 64 + col];
    b.y = W[(kk + 1) * 64 + col];
    c = wmma_f32_k4(a, b, c);
  }
  const float bb = bias[col];
#pragma unroll
  for (int r = 0; r < 8; ++r) {
    const int row = r + hf * 8;
    Y[(long long)(r0 + row) * 64 + col] = actf<ACT>(c[r] + bb);
  }
}

// h_net2 = max over 3 relations of cd_rel[row]*(agg_rel @ W_rel) + b_rel
__global__ void k_net2(const float* __restrict__ aggP, const float* __restrict__ aggF,
                       const float* __restrict__ aggS,
                       const float* __restrict__ Wp, const float* __restrict__ bp,
                       const float* __restrict__ Wf, const float* __restrict__ bf,
                       const float* __restrict__ Ws, const float* __restrict__ bs,
                       const float* __restrict__ cdP, const float* __restrict__ cdF,
                       const float* __restrict__ cdS, float* __restrict__ Y) {
  __shared__ __align__(16) float sp[16 * 64], sf[16 * 64], ss[16 * 64];
  __shared__ float scd[48];
  const int t = threadIdx.x;
  const int r0 = blockIdx.x * 16;
  const long long base = (long long)r0 * 64;
  async_g2l_b128(sp + t * 4, aggP + base + t * 4);
  async_g2l_b128(sp + 512 + t * 4, aggP + base + 512 + t * 4);
  async_g2l_b128(sf + t * 4, aggF + base + t * 4);
  async_g2l_b128(sf + 512 + t * 4, aggF + base + 512 + t * 4);
  async_g2l_b128(ss + t * 4, aggS + base + t * 4);
  async_g2l_b128(ss + 512 + t * 4, aggS + base + 512 + t * 4);
  if (t < 16) scd[t] = cdP[r0 + t];
  else if (t < 32) scd[t] = cdF[r0 + t - 16];
  else if (t < 48) scd[t] = cdS[r0 + t - 32];
  wait_async0();
  __syncthreads();
  const int lane = t & 31, wv = t >> 5, hf = lane >> 4, lm = lane & 15;
  const int col = wv * 16 + lm;
  v8f c1 = {}, c2 = {}, c3 = {};
#pragma unroll 4
  for (int kb = 0; kb < 64; kb += 4) {
    const int kk = kb + hf * 2;
    v2f a1 = *(const v2f*)(sp + lm * 64 + kk);
    v2f a2 = *(const v2f*)(sf + lm * 64 + kk);
    v2f a3 = *(const v2f*)(ss + lm * 64 + kk);
    v2f b1, b2, b3;
    b1.x = Wp[(kk + 0) * 64 + col]; b1.y = Wp[(kk + 1) * 64 + col];
    b2.x = Wf[(kk + 0) * 64 + col]; b2.y = Wf[(kk + 1) * 64 + col];
    b3.x = Ws[(kk + 0) * 64 + col]; b3.y = Ws[(kk + 1) * 64 + col];
    c1 = wmma_f32_k4(a1, b1, c1);
    c2 = wmma_f32_k4(a2, b2, c2);
    c3 = wmma_f32_k4(a3, b3, c3);
  }
  const float bbp = bp[col], bbf = bf[col], bbs = bs[col];
#pragma unroll
  for (int r = 0; r < 8; ++r) {
    const int row = r + hf * 8;
    float v1 = c1[r] * scd[row] + bbp;
    float v2 = c2[r] * scd[16 + row] + bbf;
    float v3 = c3[r] * scd[32 + row] + bbs;
    Y[(long long)(r0 + row) * 64 + col] = fmaxf(fmaxf(v1, v2), v3);
  }
}

// Fused CFConv: e = ssp(ssp(pf@W1+b1)@W2+b2); acc[pin_cell] += hv[pin_net]*e
__global__ void k_cfconv(const float* __restrict__ pf,
                         const float* __restrict__ W1, const float* __restrict__ b1,
                         const float* __restrict__ W2, const float* __restrict__ b2,
                         const float* __restrict__ hv,
                         const int* __restrict__ pin_net, const int* __restrict__ pin_cell,
                         float* __restrict__ acc) {
  __shared__ __align__(16) float st[16 * 64];
  __shared__ __align__(16) float spf[16 * 8];
  __shared__ int snet[16], scell[16];
  const int t = threadIdx.x;
  const int p0 = blockIdx.x * 16;
  if (t < 32) async_g2l_b128(spf + t * 4, pf + (long long)p0 * 8 + t * 4);
  if (t < 16) snet[t] = pin_net[p0 + t];
  else if (t < 32) scell[t - 16] = pin_cell[p0 + t - 16];
  wait_async0();
  __syncthreads();
  const int lane = t & 31, wv = t >> 5, hf = lane >> 4, lm = lane & 15;
  const int col = wv * 16 + lm;
  // stage 1: t1 tile = ssp(pin_feat[16x8] @ W1[8x64] + b1) -> LDS
  v8f c = {};
#pragma unroll
  for (int kb = 0; kb < 8; kb += 4) {
    const int kk = kb + hf * 2;
    v2f a = *(const v2f*)(spf + lm * 8 + kk);
    v2f b;
    b.x = W1[(kk + 0) * 64 + col];
    b.y = W1[(kk + 1) * 64 + col];
    c = wmma_f32_k4(a, b, c);
  }
  const float bb1 = b1[col];
#pragma unroll
  for (int r = 0; r < 8; ++r) {
    const int row = r + hf * 8;
    st[row * 64 + col] = sspf(c[r] + bb1);
  }
  __syncthreads();
  // stage 2: e tile = ssp(t1[16x64] @ W2[64x64] + b2)
  v8f c2v = {};
#pragma unroll 4
  for (int kb = 0; kb < 64; kb += 4) {
    const int kk = kb + hf * 2;
    v2f a = *(const v2f*)(st + lm * 64 + kk);
    v2f b;
    b.x = W2[(kk + 0) * 64 + col];
    b.y = W2[(kk + 1) * 64 + col];
    c2v = wmma_f32_k4(a, b, c2v);
  }
  const float bb2 = b2[col];
#pragma unroll
  for (int r = 0; r < 8; ++r) {
    const int row = r + hf * 8;
    const float e = sspf(c2v[r] + bb2);
    const int net = snet[row], cell = scell[row];
    const float val = e * hv[(long long)net * 64 + col];
    atomicAdd(&acc[(long long)cell * 64 + col], val);
  }
}

// ---------------------------------------------------------------------------
__global__ void k_net_readout(const float* __restrict__ h2, const int* __restrict__ pairs,
                              const float* __restrict__ wd, const float* __restrict__ bd,
                              const float* __restrict__ wa, const float* __restrict__ ba,
                              float* __restrict__ out_dis, float* __restrict__ out_ang,
                              int n) {
  int p = blockIdx.x * blockDim.x + threadIdx.x;
  if (p >= n) return;
  const int i = pairs[2 * p], j = pairs[2 * p + 1];
  float d = bd[0], a = ba[0];
  const float* hi = h2 + (long long)i * 64;
  const float* hj = h2 + (long long)j * 64;
#pragma unroll 4
  for (int f = 0; f < 64; ++f) {
    const float xi = hi[f], xj = hj[f];
    d += xi * wd[f] + xj * wd[64 + f];
    a += xi * wa[f] + xj * wa[64 + f];
  }
  out_dis[p] = softplusf(d);
  out_ang[p] = a;
}

__global__ void k_pin_readout(const float* __restrict__ h2, const float* __restrict__ hp,
                              const float* __restrict__ hc2,
                              const int* __restrict__ pin_net, const int* __restrict__ pin_cell,
                              const float* __restrict__ wd, const float* __restrict__ bd,
                              const float* __restrict__ wa, const float* __restrict__ ba,
                              float* __restrict__ out_dis, float* __restrict__ out_ang,
                              int n) {
  int p = blockIdx.x * blockDim.x + threadIdx.x;
  if (p >= n) return;
  const int net = pin_net[p], cell = pin_cell[p];
  float d = bd[0], a = ba[0];
  const float* hn = h2 + (long long)net * 64;
  const float* hc = hc2 + (long long)cell * 64;
#pragma unroll 4
  for (int f = 0; f < 64; ++f) { const float x = hn[f]; d += x * wd[f];       a += x * wa[f]; }
#pragma unroll
  for (int f = 0; f < 8;  ++f) { const float x = hp[(long long)p * 8 + f];
                                 d += x * wd[64 + f];  a += x * wa[64 + f]; }
#pragma unroll 4
  for (int f = 0; f < 64; ++f) { const float x = hc[f]; d += x * wd[72 + f];  a += x * wa[72 + f]; }
  out_dis[p] = softplusf(d);
  out_ang[p] = a;
}

// ---------------------------------------------------------------------------
extern "C" void kernel_launch(void* const* d_in, const int* in_sizes, int n_in,
                              void* d_out, int out_size, void* d_ws, size_t ws_size,
                              hipStream_t stream) {
  const float* cell_feat  = (const float*)d_in[0];
  const float* net_feat   = (const float*)d_in[1];
  const float* pin_feat   = (const float*)d_in[2];
  const float* cell_lin_w = (const float*)d_in[3];
  const float* cell_lin_b = (const float*)d_in[4];
  const float* net_lin_w  = (const float*)d_in[5];
  const float* net_lin_b  = (const float*)d_in[6];
  const float* pin_lin_w  = (const float*)d_in[7];
  const float* pin_lin_b  = (const float*)d_in[8];
  const float* pins_w     = (const float*)d_in[9];
  const float* pins_b     = (const float*)d_in[10];
  const float* father_w   = (const float*)d_in[11];
  const float* father_b   = (const float*)d_in[12];
  const float* son_w      = (const float*)d_in[13];
  const float* son_b      = (const float*)d_in[14];
  const float* cf_node_w  = (const float*)d_in[15];
  const float* cf_node_b  = (const float*)d_in[16];
  const float* cf_e1_w    = (const float*)d_in[17];
  const float* cf_e1_b    = (const float*)d_in[18];
  const float* cf_e2_w    = (const float*)d_in[19];
  const float* cf_e2_b    = (const float*)d_in[20];
  const float* cf_out_w   = (const float*)d_in[21];
  const float* cf_out_b   = (const float*)d_in[22];
  const float* net_dis_w  = (const float*)d_in[23];
  const float* net_dis_b  = (const float*)d_in[24];
  const float* net_ang_w  = (const float*)d_in[25];
  const float* net_ang_b  = (const float*)d_in[26];
  const float* pin_dis_w  = (const float*)d_in[27];
  const float* pin_dis_b  = (const float*)d_in[28];
  const float* pin_ang_w  = (const float*)d_in[29];
  const float* pin_ang_b  = (const float*)d_in[30];
  const int* pin_net  = (const int*)d_in[31];
  const int* pin_cell = (const int*)d_in[32];
  const int* nn_src   = (const int*)d_in[33];
  const int* nn_dst   = (const int*)d_in[34];
  const int* pairs    = (const int*)d_in[35];

  float* W = (float*)d_ws;
  float* deg_pc   = W;                      // 100000  (cells: pin out-degree)
  float* deg_pn   = W + 100000;             // 100000  (nets : pin in-degree)
  float* deg_src  = W + 200000;             // 100000  (nets : nn_src count)
  float* deg_dst  = W + 300000;             // 100000  (nets : nn_dst count)
  float* agg_pins = W + 400000;             // 6.4M
  float* agg_fa   = agg_pins + 6400000;     // 6.4M
  float* agg_so   = agg_fa + 6400000;       // 6.4M
  float* cell_acc = agg_so + 6400000;       // 6.4M     end of zeroed prefix: 26M
  float* h_cell   = cell_acc + 6400000;     // 6.4M
  float* h_net    = h_cell + 6400000;       // 6.4M
  float* h_pin    = h_net + 6400000;        // 4.0M
  float* hv       = h_pin + 4000000;        // 6.4M
  float* h_net2   = hv + 6400000;           // 6.4M
  float* h_cell2  = h_net2 + 6400000;       // 6.4M -> total 62M floats = 248MB

  float* out = (float*)d_out;
  float* out_net_dis = out;
  float* out_net_ang = out + N_PAIR;
  float* out_pin_dis = out + 2 * N_PAIR;
  float* out_pin_ang = out + 2 * N_PAIR + N_PIN;

  // 0) zero all accumulators + degrees
  k_zero<<<2048, 256, 0, stream>>>(W, 26000000LL);

  // 1) degrees -> rsqrt(clip(deg,1))
  k_count<<<(N_PIN + 255) / 256, 256, 0, stream>>>(pin_cell, deg_pc, N_PIN);
  k_count<<<(N_PIN + 255) / 256, 256, 0, stream>>>(pin_net, deg_pn, N_PIN);
  k_count<<<(N_NN + 255) / 256, 256, 0, stream>>>(nn_src, deg_src, N_NN);
  k_count<<<(N_NN + 255) / 256, 256, 0, stream>>>(nn_dst, deg_dst, N_NN);
  k_invsqrt_clip<<<(400000 + 255) / 256, 256, 0, stream>>>(W, 400000);

  // 2) node embeddings (WMMA f32 16x16x4)
  k_embed16<<<N_CELL / 16, 128, 0, stream>>>(cell_feat, cell_lin_w, cell_lin_b, h_cell);
  k_embed16<<<N_NET / 16, 128, 0, stream>>>(net_feat, net_lin_w, net_lin_b, h_net);
  k_pin_embed<<<(N_PIN + 255) / 256, 256, 0, stream>>>(pin_feat, pin_lin_w, pin_lin_b,
                                                       h_pin, N_PIN);

  // 3) GraphConv message scatter (cs applied at source)
  k_scatter_norm<<<(int)(((long long)N_PIN * 64 + 255) / 256), 256, 0, stream>>>(
      h_cell, pin_cell, pin_net, deg_pc, agg_pins, N_PIN);
  k_scatter_norm<<<(int)(((long long)N_NN * 64 + 255) / 256), 256, 0, stream>>>(
      h_net, nn_src, nn_dst, deg_src, agg_fa, N_NN);
  k_scatter_norm<<<(int)(((long long)N_NN * 64 + 255) / 256), 256, 0, stream>>>(
      h_net, nn_dst, nn_src, deg_dst, agg_so, N_NN);

  // 4) hv = h_net @ cf_node_w + b  (linear)
  k_gemm64<ACT_NONE><<<N_NET / 16, 128, 0, stream>>>(h_net, cf_node_w, cf_node_b, hv);

  // 5) h_net2 = max of three relation GEMMs (cd applied post-WMMA)
  k_net2<<<N_NET / 16, 128, 0, stream>>>(agg_pins, agg_fa, agg_so,
                                         pins_w, pins_b, father_w, father_b,
                                         son_w, son_b,
                                         deg_pn, deg_dst, deg_src, h_net2);

  // 6) fused CFConv edge network + scatter into cell_acc
  k_cfconv<<<N_PIN / 16, 128, 0, stream>>>(pin_feat, cf_e1_w, cf_e1_b, cf_e2_w,
                                           cf_e2_b, hv, pin_net, pin_cell, cell_acc);

  // 7) h_cell2 = ssp(cell_acc @ cf_out_w + b)
  k_gemm64<ACT_SSP><<<N_CELL / 16, 128, 0, stream>>>(cell_acc, cf_out_w, cf_out_b,
                                                     h_cell2);

  // 8) readouts
  k_net_readout<<<(N_PAIR + 255) / 256, 256, 0, stream>>>(
      h_net2, pairs, net_dis_w, net_dis_b, net_ang_w, net_ang_b,
      out_net_dis, out_net_ang, N_PAIR);
  k_pin_readout<<<(N_PIN + 255) / 256, 256, 0, stream>>>(
      h_net2, h_pin, h_cell2, pin_net, pin_cell,
      pin_dis_w, pin_dis_b, pin_ang_w, pin_ang_b,
      out_pin_dis, out_pin_ang, N_PIN);

  (void)in_sizes; (void)n_in; (void)out_size; (void)ws_size;
}